// VectorQuantizer_39127152067278
// MI455X (gfx1250) — compile-verified
//
#include <hip/hip_runtime.h>

typedef __attribute__((ext_vector_type(2))) float v2f;
typedef __attribute__((ext_vector_type(8))) float v8f;

#define K_ENTRIES 1024
#define DIM 64
#define CHUNK 128           // codebook entries staged per LDS pass
#define ROW_STRIDE 70       // 64 data + 1 norm + 3 zero + 2 pad: even (b64-aligned), bank-conflict-free
#define ROWS_PER_BLOCK 128  // 8 waves * 16 rows
#define N_ROWS 65536
#define QSIZE (N_ROWS * DIM)   // 4194304 floats of quantized output
// d_out layout: [ quantized : QSIZE ][ loss : 1 ][ indices(float) : N_ROWS ]

// ---------------------------------------------------------------------------
// Kernel 1: fused distance-matmul + argmin via V_WMMA_F32_16X16X4_F32.
// A-matrix  = codebook tile, staged in LDS as (-2*e) with augmented 17th
//             K-chunk {||e||^2, 0, 0, 0}  -> WMMA output IS the argmin score.
// B-matrix  = input rows (held in registers whole kernel) with augmented
//             chunk {1,0,0,0}.
// Each wave owns 16 input rows; block covers 128 rows vs all 1024 entries.
// ---------------------------------------------------------------------------
__global__ __launch_bounds__(256)
void vq_argmin_kernel(const float* __restrict__ x,
                      const float* __restrict__ cb,
                      float* __restrict__ out) {
    __shared__ float lds[CHUNK * ROW_STRIDE];   // 35,840 B

    const int tid    = threadIdx.x;
    const int lane   = tid & 31;
    const int wave   = tid >> 5;
    const int laneHi = lane >> 4;   // 0: lanes 0-15, 1: lanes 16-31
    const int l16    = lane & 15;
    const int rowbase = blockIdx.x * ROWS_PER_BLOCK + wave * 16;

    if (blockIdx.x == 0 && tid == 0) out[QSIZE] = 0.0f;  // zero loss accumulator

    // ---- load B (x tile) into registers, WMMA B layout:
    // lane l, vgpr v: n = l%16 (row), k = 2v + l/16 (dim within chunk)
    v2f B[17];
    {
        const int row = rowbase + l16;
        #pragma unroll
        for (int c = 0; c < 16; ++c) {
            const int d = 4 * c + laneHi;
            B[c].x = x[row * DIM + d];
            B[c].y = x[row * DIM + d + 2];
        }
        B[16].x = (laneHi == 0) ? 1.0f : 0.0f;  // augmented dim picks up ||e||^2
        B[16].y = 0.0f;
    }

    float minv = 3.4e38f;
    int   mini = 0;

    for (int ch = 0; ch < K_ENTRIES / CHUNK; ++ch) {
        __syncthreads();  // previous pass' compute done before LDS overwrite

        // ---- stage chunk: 128 entries x 64 dims, store -2*e (row-major, padded)
        {
            const float* src = cb + ch * CHUNK * DIM;
            if (ch + 1 < K_ENTRIES / CHUNK)
                __builtin_prefetch(cb + (ch + 1) * CHUNK * DIM + tid, 0, 0);
            #pragma unroll
            for (int t = 0; t < (CHUNK * DIM) / 256; ++t) {
                const int idx = tid + t * 256;     // 0..8191
                const int e = idx >> 6;
                const int d = idx & 63;
                lds[e * ROW_STRIDE + d] = -2.0f * src[idx];
            }
        }
        __syncthreads();

        // ---- augmented column: ||e||^2 = sum((-2e)^2)/4, then zeros
        if (tid < CHUNK) {
            float s = 0.0f;
            #pragma unroll
            for (int d = 0; d < DIM; ++d) {
                const float v = lds[tid * ROW_STRIDE + d];
                s += v * v;
            }
            lds[tid * ROW_STRIDE + 64] = 0.25f * s;
            lds[tid * ROW_STRIDE + 65] = 0.0f;
            lds[tid * ROW_STRIDE + 66] = 0.0f;
            lds[tid * ROW_STRIDE + 67] = 0.0f;
        }
        __syncthreads();

        // ---- 8 column tiles of 16 entries: 17 chained WMMAs each
        #pragma unroll
        for (int j = 0; j < CHUNK / 16; ++j) {
            const int ebase = j * 16;
            v8f acc = {};
            #pragma unroll
            for (int c = 0; c < 17; ++c) {
                // A layout: lane l, vgpr v: m = l%16 (entry), k = 2*(l/16) + v
                const int d0 = 4 * c + 2 * laneHi;
                const v2f A = *(const v2f*)&lds[(ebase + l16) * ROW_STRIDE + d0];
                acc = __builtin_amdgcn_wmma_f32_16x16x4_f32(
                        /*neg_a=*/false, A, /*neg_b=*/false, B[c],
                        /*c_mod=*/(short)0, acc,
                        /*reuse_a=*/false, /*reuse_b=*/false);
            }
            // acc[r] = score for entry (ch*CHUNK + ebase + r + 8*laneHi), row l16
            const int idx0 = ch * CHUNK + ebase + 8 * laneHi;
            #pragma unroll
            for (int r = 0; r < 8; ++r) {
                const float s = acc[r];
                const bool p = s < minv;
                minv = p ? s : minv;
                mini = p ? (idx0 + r) : mini;
            }
        }
    }

    // ---- fold the two 16-lane halves, lanes 0-15 write the 16 row indices
    {
        const float ov = __shfl_xor(minv, 16, 32);
        const int   oi = __shfl_xor(mini, 16, 32);
        if (ov < minv) { minv = ov; mini = oi; }
    }
    if (laneHi == 0) {
        out[QSIZE + 1 + rowbase + l16] = (float)mini;
    }
}

// ---------------------------------------------------------------------------
// Kernel 2: gather codebook[idx], straight-through output x + (q - x),
// loss = (1 + 0.25) * mean((q - x)^2)   (e-loss == q-loss numerically).
// ---------------------------------------------------------------------------
__global__ __launch_bounds__(256)
void vq_gather_loss_kernel(const float* __restrict__ x,
                           const float* __restrict__ cb,
                           float* __restrict__ out) {
    const float* idxf = out + QSIZE + 1;
    float* loss = out + QSIZE;

    const int tid = threadIdx.x;
    const long base = (long)blockIdx.x * 4096;
    float acc = 0.0f;

    #pragma unroll
    for (int t = 0; t < 16; ++t) {
        const long i = base + t * 256 + tid;
        const int row = (int)(i >> 6);
        const int d   = (int)(i & 63);
        const int k   = (int)idxf[row];
        const float q  = cb[k * DIM + d];
        const float xv = x[i];
        out[i] = xv + (q - xv);           // straight-through, bit-matches ref
        const float diff = q - xv;
        acc += diff * diff;
    }

    // block reduction: wave32 shuffle tree + LDS across 8 waves
    for (int m = 16; m >= 1; m >>= 1) acc += __shfl_xor(acc, m, 32);
    __shared__ float wsum[8];
    if ((tid & 31) == 0) wsum[tid >> 5] = acc;
    __syncthreads();
    if (tid < 8) {
        float s = wsum[tid];
        for (int m = 4; m >= 1; m >>= 1) s += __shfl_xor(s, m, 32);
        if (tid == 0) atomicAdd(loss, s * (1.25f / (float)QSIZE));
    }
}

extern "C" void kernel_launch(void* const* d_in, const int* in_sizes, int n_in,
                              void* d_out, int out_size, void* d_ws, size_t ws_size,
                              hipStream_t stream) {
    (void)in_sizes; (void)n_in; (void)out_size; (void)d_ws; (void)ws_size;
    const float* x  = (const float*)d_in[0];   // [64,32,32,64] fp32
    const float* cb = (const float*)d_in[1];   // [1024,64]    fp32
    float* out = (float*)d_out;

    vq_argmin_kernel<<<N_ROWS / ROWS_PER_BLOCK, 256, 0, stream>>>(x, cb, out);
    vq_gather_loss_kernel<<<QSIZE / 4096, 256, 0, stream>>>(x, cb, out);
}